// HybridAttention_3745211482984
// MI455X (gfx1250) — compile-verified
//
#include <hip/hip_runtime.h>
#include <hip/hip_bf16.h>

typedef __attribute__((ext_vector_type(2))) float v2f;
typedef __attribute__((ext_vector_type(8))) float v8f;

#define Bsz 64
#define Lsz 256
#define HID 512
#define NH 8
#define Esz 64
#define NFREQ 78      // band f = 51..128
#define F0 51
#define TOPK 5
#define DTROWS 160    // 156 used, padded to 160
#define TWO_PI_OVER_L 0.0245436926061702596754894014318f  // 2*pi/256

// ---------------- WMMA f32 16x16x4 helpers ----------------
// A (16x4): lane m = lane&15 ; VGPR0/1 hold K = k0+2*(lane>>4), +1
__device__ __forceinline__ v2f load_a4(const float* base, int stride, int m0, int k0, int lane) {
  int m = m0 + (lane & 15);
  int kb = k0 + ((lane >> 4) << 1);
  v2f a;
  a.x = base[m * stride + kb];
  a.y = base[m * stride + kb + 1];
  return a;
}
// B (4x16): lane n = lane&15 ; VGPR0/1 hold K = k0+2*(lane>>4), +1
__device__ __forceinline__ v2f load_b4(const float* base, int stride, int k0, int n0, int lane) {
  int n = n0 + (lane & 15);
  int kb = k0 + ((lane >> 4) << 1);
  v2f b;
  b.x = base[kb * stride + n];
  b.y = base[(kb + 1) * stride + n];
  return b;
}
__device__ __forceinline__ v8f wmma4(v2f a, v2f b, v8f c) {
  return __builtin_amdgcn_wmma_f32_16x16x4_f32(false, a, false, b, (short)0, c, false, false);
}

// ---------------- Generic f32 WMMA GEMM: C = A(MxK) @ B(KxN) (+bias) ----------------
// Row-major. Tile 64x64, 256 threads (8 waves), K staged in chunks of 32.
// Requires K % 32 == 0, N % 64 == 0. M guarded.
__global__ __launch_bounds__(256) void gemm_f32_kernel(
    const float* __restrict__ A, const float* __restrict__ B, float* __restrict__ C,
    const float* __restrict__ bias, int M, int N, int K, int lda, int ldb, int ldc,
    long long sA, long long sB, long long sC) {
  __shared__ float As[64][33];  // padded: A-fragment loads are column accesses
  __shared__ float Bs[32][64];  // B-fragment loads are row accesses
  const int bz = blockIdx.z;
  const float* Ab = A + (long long)bz * sA;
  const float* Bb = B + (long long)bz * sB;
  float* Cb = C + (long long)bz * sC;
  const int m0 = blockIdx.y * 64;
  const int n0 = blockIdx.x * 64;
  const int tid = threadIdx.x;
  const int lane = tid & 31;
  const int wave = tid >> 5;   // 0..7
  const int mt = wave >> 1;    // m-tile 0..3
  const int nh = wave & 1;     // n-half 0..1

  v8f acc0 = {};
  v8f acc1 = {};

  for (int kc = 0; kc < K; kc += 32) {
    for (int i = tid; i < 64 * 32; i += 256) {
      int r = i >> 5, c = i & 31;
      int gm = m0 + r;
      As[r][c] = (gm < M) ? Ab[(long long)gm * lda + (kc + c)] : 0.0f;
    }
    for (int i = tid; i < 32 * 64; i += 256) {
      int r = i >> 6, c = i & 63;
      Bs[r][c] = Bb[(long long)(kc + r) * ldb + (n0 + c)];
    }
    __syncthreads();
#pragma unroll
    for (int k0 = 0; k0 < 32; k0 += 4) {
      v2f a  = load_a4(&As[0][0], 33, mt * 16, k0, lane);
      v2f b0 = load_b4(&Bs[0][0], 64, k0, nh * 32, lane);
      v2f b1 = load_b4(&Bs[0][0], 64, k0, nh * 32 + 16, lane);
      acc0 = wmma4(a, b0, acc0);
      acc1 = wmma4(a, b1, acc1);
    }
    __syncthreads();
  }

  const int col = lane & 15;
  const int rofs = (lane >> 4) << 3;
#pragma unroll
  for (int t = 0; t < 2; t++) {
    v8f acc = t ? acc1 : acc0;
    int nn = n0 + nh * 32 + t * 16 + col;
    float bv = bias ? bias[nn] : 0.0f;
#pragma unroll
    for (int v = 0; v < 8; v++) {
      int mm = m0 + mt * 16 + v + rofs;
      if (mm < M) Cb[(long long)mm * ldc + nn] = acc[v] + bv;
    }
  }
}

// ---------------- Build band DFT matrix DT (160 x 256) ----------------
// row 2j   =  cos(2*pi*(51+j)*t/256)   (Re of rfft)
// row 2j+1 = -sin(2*pi*(51+j)*t/256)   (Im of rfft)
__global__ void dft_kernel(float* __restrict__ DT) {
  int r = blockIdx.x;
  int t = threadIdx.x;
  float v = 0.0f;
  if (r < 2 * NFREQ) {
    int j = r >> 1;
    int f = F0 + j;
    float th = TWO_PI_OVER_L * (float)((f * t) & 255);
    v = (r & 1) ? -sinf(th) : cosf(th);
  }
  DT[r * Lsz + t] = v;
}

// ---------------- Spectral product + channel mean: S[b,f] ----------------
// Sr = mean_c(QrKr + QiKi), Si = mean_c(QiKr - QrKi)
__global__ void sred_kernel(const float* __restrict__ Qh, const float* __restrict__ Kh,
                            float* __restrict__ Sr, float* __restrict__ Si) {
  int b = blockIdx.x / NFREQ;
  int j = blockIdx.x % NFREQ;
  const float* qr = Qh + ((long long)b * DTROWS + 2 * j) * HID;
  const float* qi = qr + HID;
  const float* kr = Kh + ((long long)b * DTROWS + 2 * j) * HID;
  const float* ki = kr + HID;
  __shared__ float s1[256], s2[256];
  int tid = threadIdx.x;
  float ar = 0.f, ai = 0.f;
  for (int c = tid; c < HID; c += 256) {
    float a = qr[c], bq = qi[c], cc = kr[c], d = ki[c];
    ar += a * cc + bq * d;
    ai += bq * cc - a * d;
  }
  s1[tid] = ar; s2[tid] = ai;
  __syncthreads();
  for (int s = 128; s > 0; s >>= 1) {
    if (tid < s) { s1[tid] += s1[tid + s]; s2[tid] += s2[tid + s]; }
    __syncthreads();
  }
  if (tid == 0) {
    Sr[b * NFREQ + j] = s1[0] * (1.0f / (float)HID);
    Si[b * NFREQ + j] = s2[0] * (1.0f / (float)HID);
  }
}

// ---------------- Band-limited irfft of S -> mean_value[b,tau] ----------------
__global__ void meanval_kernel(const float* __restrict__ Sr, const float* __restrict__ Si,
                               float* __restrict__ mv) {
  int b = blockIdx.x;
  int tau = threadIdx.x;
  const float* sr = Sr + b * NFREQ;
  const float* si = Si + b * NFREQ;
  float acc = sr[NFREQ - 1] * ((tau & 1) ? -1.0f : 1.0f);  // Nyquist f=128, weight 1
  for (int j = 0; j < NFREQ - 1; j++) {
    float th = TWO_PI_OVER_L * (float)(((F0 + j) * tau) & 255);
    acc += 2.0f * (sr[j] * cosf(th) - si[j] * sinf(th));
  }
  mv[b * Lsz + tau] = acc * (1.0f / (float)Lsz);
}

// ---------------- Top-K over batch-mean + per-batch softmax ----------------
__global__ void topk_kernel(const float* __restrict__ mv, int* __restrict__ idx_out,
                            float* __restrict__ tc_out) {
  __shared__ float g[Lsz];
  __shared__ int sidx[TOPK];
  int tid = threadIdx.x;
  float s = 0.f;
  for (int b = 0; b < Bsz; b++) s += mv[b * Lsz + tid];
  g[tid] = s;  // scale-invariant for argmax
  __syncthreads();
  if (tid == 0) {
    for (int k = 0; k < TOPK; k++) {
      int best = 0; float bv = -INFINITY;
      for (int t = 0; t < Lsz; t++) {
        if (g[t] > bv) { bv = g[t]; best = t; }  // strict > : lowest index on tie
      }
      sidx[k] = best;
      idx_out[k] = best;
      g[best] = -INFINITY;
    }
  }
  __syncthreads();
  if (tid < Bsz) {
    float w[TOPK]; float mx = -INFINITY;
    for (int k = 0; k < TOPK; k++) { w[k] = mv[tid * Lsz + sidx[k]]; mx = fmaxf(mx, w[k]); }
    float den = 0.f;
    for (int k = 0; k < TOPK; k++) { w[k] = expf(w[k] - mx); den += w[k]; }
    for (int k = 0; k < TOPK; k++) tc_out[tid * TOPK + k] = w[k] / den;
  }
}

// ---------------- Frequency-domain context: rolled weighted sum of V ----------------
__global__ void ctime_kernel(const float* __restrict__ V, const int* __restrict__ idx,
                             const float* __restrict__ tc, float* __restrict__ ct) {
  long long bl = blockIdx.x;
  int b = (int)(bl >> 8);
  int l = (int)(bl & 255);
  __shared__ int sIdx[TOPK];
  __shared__ float sTc[TOPK];
  if (threadIdx.x < TOPK) { sIdx[threadIdx.x] = idx[threadIdx.x]; sTc[threadIdx.x] = tc[b * TOPK + threadIdx.x]; }
  __syncthreads();
  for (int c = threadIdx.x; c < HID; c += 256) {
    float acc = 0.f;
#pragma unroll
    for (int k = 0; k < TOPK; k++) {
      int src = (l + sIdx[k]) & 255;
      acc += sTc[k] * V[((long long)b * Lsz + src) * HID + c];
    }
    ct[bl * HID + c] = acc;
  }
}

// ---------------- Flash attention per (b,h); fuses 0.9*freq + 0.1*spatial ----------------
#define ACH 32
__global__ __launch_bounds__(128) void attn_kernel(
    const float* __restrict__ Q, const float* __restrict__ K, const float* __restrict__ V,
    const float* __restrict__ mask, float* __restrict__ ctx) {
  int bh = blockIdx.x;
  int b = bh >> 3, h = bh & 7;
  int ql0 = blockIdx.y * 64;
  int tid = threadIdx.x;
  int lane = tid & 31;
  int wave = tid >> 5;  // 0..3, each owns 16 query rows

  __shared__ float Qt[64][65];        // [l][e]   (A loads -> padded)
  __shared__ float Ks[Esz][ACH];      // [e][s]   (B loads -> row access)
  __shared__ float Vs[ACH][Esz];      // [s][e]   (B loads -> row access)
  __shared__ float Ps[4][16][33];     // per-wave P tile (A loads -> padded)

  for (int i = tid; i < 64 * 64; i += 128) {
    int r = i >> 6, c = i & 63;
    Qt[r][c] = Q[((long long)(b * Lsz + ql0 + r)) * HID + h * Esz + c];
  }

  v8f O0 = {}, O1 = {}, O2 = {}, O3 = {};
  float m_i[8], l_i[8];
#pragma unroll
  for (int v = 0; v < 8; v++) { m_i[v] = -INFINITY; l_i[v] = 0.f; }

  const int col = lane & 15;
  const int rofs = (lane >> 4) << 3;
  const float inv_sqrtE = 0.125f;

  for (int sc = 0; sc < Lsz / ACH; sc++) {
    int s0 = sc * ACH;
    for (int i = tid; i < ACH * Esz; i += 128) {
      int s = i >> 6, e = i & 63;
      long long gofs = ((long long)(b * Lsz + s0 + s)) * HID + h * Esz + e;
      Ks[e][s] = K[gofs];
      Vs[s][e] = V[gofs];
    }
    __syncthreads();

    // S = Q_tile(16x64) @ K_chunk^T  -> 16 x 32
    v8f st0 = {}, st1 = {};
#pragma unroll
    for (int k0 = 0; k0 < Esz; k0 += 4) {
      v2f a  = load_a4(&Qt[0][0], 65, wave * 16, k0, lane);
      v2f b0 = load_b4(&Ks[0][0], ACH, k0, 0, lane);
      v2f b1 = load_b4(&Ks[0][0], ACH, k0, 16, lane);
      st0 = wmma4(a, b0, st0);
      st1 = wmma4(a, b1, st1);
    }

    // scale + mask + online softmax (rows live in half-waves; width-16 reductions)
#pragma unroll
    for (int v = 0; v < 8; v++) {
      int grow = ql0 + wave * 16 + v + rofs;
      long long mbase = ((long long)b * Lsz + grow) * Lsz + s0;
      float x0 = st0[v] * inv_sqrtE + mask[mbase + col];
      float x1 = st1[v] * inv_sqrtE + mask[mbase + 16 + col];
      float mx = fmaxf(x0, x1);
      mx = fmaxf(mx, __shfl_xor(mx, 1, 16));
      mx = fmaxf(mx, __shfl_xor(mx, 2, 16));
      mx = fmaxf(mx, __shfl_xor(mx, 4, 16));
      mx = fmaxf(mx, __shfl_xor(mx, 8, 16));
      float mnew = fmaxf(m_i[v], mx);
      float scl = (m_i[v] == -INFINITY) ? 0.0f : expf(m_i[v] - mnew);
      float p0 = expf(x0 - mnew);
      float p1 = expf(x1 - mnew);
      float rs = p0 + p1;
      rs += __shfl_xor(rs, 1, 16);
      rs += __shfl_xor(rs, 2, 16);
      rs += __shfl_xor(rs, 4, 16);
      rs += __shfl_xor(rs, 8, 16);
      l_i[v] = l_i[v] * scl + rs;
      m_i[v] = mnew;
      O0[v] *= scl; O1[v] *= scl; O2[v] *= scl; O3[v] *= scl;
      Ps[wave][v + rofs][col] = p0;
      Ps[wave][v + rofs][16 + col] = p1;
    }

    // O += P(16x32) @ V_chunk(32x64)   (DS ops in-order within wave: no barrier needed)
#pragma unroll
    for (int k0 = 0; k0 < ACH; k0 += 4) {
      v2f a  = load_a4(&Ps[wave][0][0], 33, 0, k0, lane);
      v2f b0 = load_b4(&Vs[0][0], Esz, k0, 0, lane);
      v2f b1 = load_b4(&Vs[0][0], Esz, k0, 16, lane);
      v2f b2 = load_b4(&Vs[0][0], Esz, k0, 32, lane);
      v2f b3 = load_b4(&Vs[0][0], Esz, k0, 48, lane);
      O0 = wmma4(a, b0, O0);
      O1 = wmma4(a, b1, O1);
      O2 = wmma4(a, b2, O2);
      O3 = wmma4(a, b3, O3);
    }
    __syncthreads();
  }

  // epilogue: normalize and blend into frequency context (in place)
#pragma unroll
  for (int et = 0; et < 4; et++) {
    v8f O = (et == 0) ? O0 : (et == 1) ? O1 : (et == 2) ? O2 : O3;
#pragma unroll
    for (int v = 0; v < 8; v++) {
      int row = ql0 + wave * 16 + v + rofs;
      int e = et * 16 + col;
      long long o = ((long long)(b * Lsz + row)) * HID + h * Esz + e;
      float val = O[v] / l_i[v];
      ctx[o] = 0.9f * ctx[o] + 0.1f * val;
    }
  }
}

// ---------------- Residual + LayerNorm ----------------
__global__ void ln_kernel(const float* __restrict__ hidden, const float* __restrict__ x_in,
                          const float* __restrict__ gamma, const float* __restrict__ beta,
                          float* __restrict__ out) {
  long long row = blockIdx.x;
  const float* hp = hidden + row * HID;
  const float* xp = x_in + row * HID;
  __shared__ float s1[256], s2[256];
  int tid = threadIdx.x;
  float a0 = hp[tid] + xp[tid];
  float a1 = hp[tid + 256] + xp[tid + 256];
  s1[tid] = a0 + a1;
  s2[tid] = a0 * a0 + a1 * a1;
  __syncthreads();
  for (int s = 128; s > 0; s >>= 1) {
    if (tid < s) { s1[tid] += s1[tid + s]; s2[tid] += s2[tid + s]; }
    __syncthreads();
  }
  float mu = s1[0] * (1.0f / (float)HID);
  float var = s2[0] * (1.0f / (float)HID) - mu * mu;
  float inv = rsqrtf(var + 1e-12f);
  out[row * HID + tid] = (a0 - mu) * inv * gamma[tid] + beta[tid];
  out[row * HID + tid + 256] = (a1 - mu) * inv * gamma[tid + 256] + beta[tid + 256];
}

// ---------------- Host launcher ----------------
extern "C" void kernel_launch(void* const* d_in, const int* in_sizes, int n_in,
                              void* d_out, int out_size, void* d_ws, size_t ws_size,
                              hipStream_t stream) {
  const float* X    = (const float*)d_in[0];   // (B,L,HID)
  const float* Msk  = (const float*)d_in[1];   // (B,1,L,L)
  const float* Wq   = (const float*)d_in[2];
  const float* bq   = (const float*)d_in[3];
  const float* Wk   = (const float*)d_in[4];
  const float* bk   = (const float*)d_in[5];
  const float* Wv   = (const float*)d_in[6];
  const float* bv   = (const float*)d_in[7];
  const float* Wd   = (const float*)d_in[8];
  const float* bd   = (const float*)d_in[9];
  const float* gamma = (const float*)d_in[10];
  const float* beta  = (const float*)d_in[11];
  float* out = (float*)d_out;

  const long long NROW = (long long)Bsz * Lsz;        // 16384
  const long long QKV  = NROW * HID;                  // 8388608
  float* ws = (float*)d_ws;
  float* Qb = ws;                 long long o = QKV;
  float* Kb = ws + o;             o += QKV;
  float* Vb = ws + o;             o += QKV;
  float* DT = ws + o;             o += (long long)DTROWS * Lsz;
  float* Qh = ws + o;             o += (long long)Bsz * DTROWS * HID;
  float* Kh = ws + o;             o += (long long)Bsz * DTROWS * HID;
  float* Sr = ws + o;             o += (long long)Bsz * NFREQ;
  float* Si = ws + o;             o += (long long)Bsz * NFREQ;
  float* MV = ws + o;             o += (long long)Bsz * Lsz;
  float* TC = ws + o;             o += (long long)Bsz * TOPK;
  float* CT = ws + o;             o += QKV;
  float* H  = ws + o;             o += QKV;
  int*   Idx = (int*)(ws + o);

  // 1. DFT matrix
  dft_kernel<<<DTROWS, Lsz, 0, stream>>>(DT);

  // 2. Q/K/V projections (WMMA f32)
  dim3 gBig(HID / 64, (unsigned)(NROW / 64), 1);
  gemm_f32_kernel<<<gBig, 256, 0, stream>>>(X, Wq, Qb, bq, (int)NROW, HID, HID, HID, HID, HID, 0, 0, 0);
  gemm_f32_kernel<<<gBig, 256, 0, stream>>>(X, Wk, Kb, bk, (int)NROW, HID, HID, HID, HID, HID, 0, 0, 0);
  gemm_f32_kernel<<<gBig, 256, 0, stream>>>(X, Wv, Vb, bv, (int)NROW, HID, HID, HID, HID, HID, 0, 0, 0);

  // 3. Band DFT of Q,K: per-batch (160x512) = DT(160x256) @ Q_b(256x512)
  dim3 gDft(HID / 64, (DTROWS + 63) / 64, Bsz);
  gemm_f32_kernel<<<gDft, 256, 0, stream>>>(DT, Qb, Qh, nullptr, DTROWS, HID, Lsz,
                                            Lsz, HID, HID, 0, (long long)Lsz * HID, (long long)DTROWS * HID);
  gemm_f32_kernel<<<gDft, 256, 0, stream>>>(DT, Kb, Kh, nullptr, DTROWS, HID, Lsz,
                                            Lsz, HID, HID, 0, (long long)Lsz * HID, (long long)DTROWS * HID);

  // 4. Spectral product + channel mean
  sred_kernel<<<Bsz * NFREQ, 256, 0, stream>>>(Qh, Kh, Sr, Si);

  // 5. mean_value via band-limited inverse DFT
  meanval_kernel<<<Bsz, Lsz, 0, stream>>>(Sr, Si, MV);

  // 6. top-5 indices + per-batch softmax weights
  topk_kernel<<<1, Lsz, 0, stream>>>(MV, Idx, TC);

  // 7. frequency context (rolled weighted V)
  ctime_kernel<<<(unsigned)NROW, 256, 0, stream>>>(Vb, Idx, TC, CT);

  // 8. flash attention + blend into CT (in place)
  attn_kernel<<<dim3(Bsz * NH, Lsz / 64), 128, 0, stream>>>(Qb, Kb, Vb, Msk, CT);

  // 9. output projection
  gemm_f32_kernel<<<gBig, 256, 0, stream>>>(CT, Wd, H, bd, (int)NROW, HID, HID, HID, HID, HID, 0, 0, 0);

  // 10. residual + LayerNorm
  ln_kernel<<<(unsigned)NROW, 256, 0, stream>>>(H, X, gamma, beta, out);
}